// BpMapping_10986526343937
// MI455X (gfx1250) — compile-verified
//
#include <hip/hip_runtime.h>
#include <math.h>

// ---- Problem constants (match reference) ----
#define NXI   256
#define NYI   256
#define NPIX  (NXI * NYI)          // 65536 pixels, 256 KB as f32
#define NSAMP 128
#define NEV   200000
#define NB    2
#define SPEEDF 0.3f                 // mm per ps

// ---- Launch configuration ----
#define BLOCK     1024              // 32 wave32 waves per workgroup
#define WGS_PER_B 96                // workgroups per image batch
#define NWG       (NB * WGS_PER_B)

typedef unsigned int uint4v __attribute__((ext_vector_type(4)));
typedef unsigned int uint8v __attribute__((ext_vector_type(8)));

// ============================================================================
// Fused forward-project + back-project kernel.
// Each workgroup owns ONE batch image and keeps the full 256KB back-projection
// accumulator in LDS (MI455X: 320KB LDS per WGP). Event geometry is staged
// into LDS with the Tensor Data Mover (tensor_load_to_lds, TENSORcnt); the
// tof array uses per-lane async global->LDS copies (ASYNCcnt). Scatter-adds
// go to LDS (ds_add_f32); one flush per workgroup uses global_atomic_add_f32.
// ============================================================================
__global__ __launch_bounds__(BLOCK) void pet_grad_kernel(
    const float* __restrict__ image,      // [B, NX, NY]
    const float* __restrict__ proj_data,  // [B, E]
    const float* __restrict__ tof,        // [E]
    const float* __restrict__ x1l, const float* __restrict__ y1l,
    const float* __restrict__ x1r, const float* __restrict__ y1r,
    const float* __restrict__ x2l, const float* __restrict__ y2l,
    const float* __restrict__ x2r, const float* __restrict__ y2r,
    const float* __restrict__ tres,       // [1] time resolution (ps)
    float* __restrict__ back)             // [B, NX, NY] accumulator (pre-zeroed)
{
    __shared__ float s_img[NPIX];          // 256 KB back-projection accumulator
    __shared__ float s_ev[9][BLOCK];       // 36 KB TDM/async-staged event data

    const int tid = threadIdx.x;
    const int b   = blockIdx.x % NB;       // which image batch this WG owns
    const int wg  = blockIdx.x / NB;       // WG index within this batch

    const float* __restrict__ img_b = image + (size_t)b * NPIX;

    // Zero the LDS accumulator.
    for (int p = tid; p < NPIX; p += BLOCK) s_img[p] = 0.0f;
    __syncthreads();

    const float sigma     = 0.5f * SPEEDF * tres[0];
    const float inv_sigma = 1.0f / sigma;
    const float gnorm     = 0.3989422804014327f * inv_sigma; // 1/(sqrt(2pi)*sigma)
    const float inv_ns    = 1.0f / (float)NSAMP;

    // --- Tensor Data Mover: 1D tile of BLOCK contiguous f32 -> LDS. ---
    // D# group0: count=1 | lds_addr | global_addr[56:0] | type=2.
    // D# group1: data_size=4B; tensor_dim0 = remaining events (tail reads
    // beyond the tensor return zero); tile_dim0 = BLOCK; dims 1/2 unused.
#define TDM_LOAD(k, srcptr)                                                    \
    do {                                                                       \
        unsigned long long _ga = (unsigned long long)(const void*)((srcptr) + base); \
        unsigned _lds = (unsigned)(size_t)(&s_ev[(k)][0]);                     \
        unsigned _rem = (unsigned)(NEV - base);                                \
        uint4v _g0;                                                            \
        _g0.x = 1u;                                    /* count = 1 */         \
        _g0.y = _lds;                                  /* lds_addr */          \
        _g0.z = (unsigned)_ga;                         /* global_addr lo */    \
        _g0.w = ((unsigned)(_ga >> 32) & 0x01FFFFFFu) | 0x80000000u; /* type=2 */ \
        uint8v _g1;                                                            \
        _g1.s0 = 0x00020000u;                          /* data_size = 4B */    \
        _g1.s1 = (_rem & 0xFFFFu) << 16;               /* tensor_dim0 lo16 */  \
        _g1.s2 = ((_rem >> 16) & 0xFFFFu) | (1u << 16);/* dim0 hi, dim1=1 */   \
        _g1.s3 = ((unsigned)BLOCK) << 16;              /* tile_dim0 = BLOCK */ \
        _g1.s4 = 0u;                                   /* tile_dim1/2 = 0 */   \
        _g1.s5 = _rem;                                 /* dim0_stride lo */    \
        _g1.s6 = 0u;                                                           \
        _g1.s7 = 0u;                                                           \
        asm volatile("tensor_load_to_lds %0, %1" :: "s"(_g0), "s"(_g1)         \
                     : "memory");                                              \
    } while (0)

    // --- Per-lane async global->LDS b32 copy (ASYNCcnt path). ---
#define ASYNC_LD(k, srcptr)                                                    \
    do {                                                                       \
        unsigned _off = (unsigned)(size_t)(&s_ev[(k)][tid]);                   \
        const float* _gp = (srcptr) + e;                                       \
        asm volatile("global_load_async_to_lds_b32 %0, %1, off"                \
                     :: "v"(_off), "v"(_gp) : "memory");                       \
    } while (0)

    for (int base = wg * BLOCK; base < NEV; base += WGS_PER_B * BLOCK) {
        const int e = base + tid;

        if (tid < 32) {   // wave 0 drives the TDM (per-wave DMA, EXEC-agnostic)
            TDM_LOAD(0, x1l); TDM_LOAD(1, y1l); TDM_LOAD(2, x1r); TDM_LOAD(3, y1r);
            TDM_LOAD(4, x2l); TDM_LOAD(5, y2l); TDM_LOAD(6, x2r); TDM_LOAD(7, y2r);
#if __has_builtin(__builtin_amdgcn_s_wait_tensorcnt)
            __builtin_amdgcn_s_wait_tensorcnt(0);
#else
            asm volatile("s_wait_tensorcnt 0x0" ::: "memory");
#endif
        }
        if (e < NEV) {
            ASYNC_LD(8, tof);
        }
        asm volatile("s_wait_asynccnt 0" ::: "memory");
        __syncthreads();   // staged data visible to all waves

        if (e < NEV) {
            const float ex1 = 0.5f * (s_ev[0][tid] + s_ev[2][tid]);
            const float ey1 = 0.5f * (s_ev[1][tid] + s_ev[3][tid]);
            const float ex2 = 0.5f * (s_ev[4][tid] + s_ev[6][tid]);
            const float ey2 = 0.5f * (s_ev[5][tid] + s_ev[7][tid]);
            const float tv  = s_ev[8][tid];

            const float ddx  = ex2 - ex1;
            const float ddy  = ey2 - ey1;
            const float L    = sqrtf(ddx * ddx + ddy * ddy);
            const float step = L * inv_ns;
            const float c    = step * gnorm;

            // ---- Pass 1: forward projection dot product ----
            float dot = 0.0f;
            #pragma unroll 4
            for (int s = 0; s < NSAMP; ++s) {
                const float t    = ((float)s + 0.5f) * inv_ns;
                const float px   = ex1 + t * ddx;
                const float py   = ey1 + t * ddy;
                const float smid = (t - 0.5f) * L;
                const float d    = (smid - tv) * inv_sigma;
                const float w    = c * __expf(-0.5f * d * d);
                const int   ix   = (int)floorf(px * 0.5f + 128.0f);
                const int   iy   = (int)floorf(py * 0.5f + 128.0f);
                if (ix >= 0 && ix < NXI && iy >= 0 && iy < NYI) {
                    dot += img_b[ix * NYI + iy] * w;
                }
            }

            const float diff = dot - proj_data[(size_t)b * NEV + e];

            // ---- Pass 2: scatter diff*w into the LDS accumulator ----
            #pragma unroll 4
            for (int s = 0; s < NSAMP; ++s) {
                const float t    = ((float)s + 0.5f) * inv_ns;
                const float px   = ex1 + t * ddx;
                const float py   = ey1 + t * ddy;
                const float smid = (t - 0.5f) * L;
                const float d    = (smid - tv) * inv_sigma;
                const float w    = c * __expf(-0.5f * d * d);
                const int   ix   = (int)floorf(px * 0.5f + 128.0f);
                const int   iy   = (int)floorf(py * 0.5f + 128.0f);
                if (ix >= 0 && ix < NXI && iy >= 0 && iy < NYI) {
                    atomicAdd(&s_img[ix * NYI + iy], diff * w);   // ds_add_f32
                }
            }
        }
        __syncthreads();   // reads done before next tile's DMA overwrites s_ev
    }
#undef TDM_LOAD
#undef ASYNC_LD

    // ---- Flush LDS accumulator to the global back buffer ----
    float* __restrict__ back_b = back + (size_t)b * NPIX;
    for (int p = tid; p < NPIX; p += BLOCK) {
        const float v = s_img[p];
        if (v != 0.0f) {
#ifdef __HIP_PLATFORM_AMD__
            unsafeAtomicAdd(&back_b[p], v);   // global_atomic_add_f32
#else
            atomicAdd(&back_b[p], v);
#endif
        }
    }
}

// ============================================================================
// Helpers: zero workspace, final AXPY epilogue out = image - s * back
// ============================================================================
__global__ void zero_kernel(float* __restrict__ p, int n) {
    int i = blockIdx.x * blockDim.x + threadIdx.x;
    if (i < n) p[i] = 0.0f;
}

__global__ void finish_kernel(const float* __restrict__ image,
                              const float* __restrict__ back,
                              const float* __restrict__ s_factor,
                              float* __restrict__ out, int n) {
    int i = blockIdx.x * blockDim.x + threadIdx.x;
    if (i < n) out[i] = image[i] - s_factor[0] * back[i];
}

// ============================================================================
extern "C" void kernel_launch(void* const* d_in, const int* in_sizes, int n_in,
                              void* d_out, int out_size, void* d_ws, size_t ws_size,
                              hipStream_t stream) {
    const float* s_factor = (const float*)d_in[0];   // [1]
    const float* image    = (const float*)d_in[1];   // [B,NX,NY]
    const float* proj     = (const float*)d_in[2];   // [B,E]
    const float* tof      = (const float*)d_in[3];   // [E]
    const float* x1l      = (const float*)d_in[4];
    const float* y1l      = (const float*)d_in[5];
    const float* x1r      = (const float*)d_in[6];
    const float* y1r      = (const float*)d_in[7];
    const float* x2l      = (const float*)d_in[8];
    const float* y2l      = (const float*)d_in[9];
    const float* x2r      = (const float*)d_in[10];
    const float* y2r      = (const float*)d_in[11];
    const float* tres     = (const float*)d_in[12];  // [1]

    float* back = (float*)d_ws;                      // B*NX*NY floats (512 KB)
    const int n = NB * NPIX;

    zero_kernel<<<(n + 255) / 256, 256, 0, stream>>>(back, n);

    pet_grad_kernel<<<NWG, BLOCK, 0, stream>>>(
        image, proj, tof, x1l, y1l, x1r, y1r, x2l, y2l, x2r, y2r, tres, back);

    finish_kernel<<<(n + 255) / 256, 256, 0, stream>>>(
        image, back, s_factor, (float*)d_out, n);
}